// SymmetricContraction_18691697672779
// MI455X (gfx1250) — compile-verified
//
#include <hip/hip_runtime.h>

// ---------------------------------------------------------------------------
// SymmetricContraction (MACE-style) for MI455X / gfx1250, wave32 + WMMA.
//
// out[b,c,o] = sum_k3 w3[e,k3,c]*G3[(o,k3),bc] + sum_k2 w2[e,k2,c]*G2[(o,k2),bc]
//            + sum_k1 w1[e,k1,c]*G1[(o,k1),bc]
// with G3 = U3f[256x4096] @ X3[4096 x batch], X3[(i,j,l)] = x_i*x_j*x_l, etc.
// All GEMMs run on V_WMMA_F32_16X16X4_F32 (fp32 in, fp32 accum).
//
// v2: 2 N-tiles (32 columns) per workgroup -> each A fragment feeds 2 B
// fragments: halves L2 A-traffic (17.7GB -> 8.9GB/launch), doubles FLOP/byte.
// ---------------------------------------------------------------------------

typedef __attribute__((ext_vector_type(2))) float v2f;
typedef __attribute__((ext_vector_type(8))) float v8f;

#define NC_D      16     // D = DOUT = 16
#define CCH       64     // channels
#define MT3       16     // M-tiles for G3 (256 rows)
#define KS3       1024   // 4096 / 4 K-steps
#define MT2       8      // 128 rows
#define KS2       64     // 256 / 4
#define MT1       4      // 64 rows
#define KS1       4      // 16 / 4
#define NT        2      // N-tiles (of 16 cols) per workgroup
#define NCOLS     (NT * 16)

// ---------------------------------------------------------------------------
// WMMA wrapper: D = A(16x4 f32) * B(4x16 f32) + C(16x16 f32)
// A frag: lane L holds M=L&15, K=(L>>4)*2+v  (v2f)
// B frag: lane L holds N=L&15, K=(L>>4)*2+v  (v2f)
// D frag: reg r holds M=r+8*(L>>4), N=L&15   (v8f)
// ---------------------------------------------------------------------------
#if __has_builtin(__builtin_amdgcn_wmma_f32_16x16x4_f32)
static __device__ __forceinline__ v8f wmma4(v2f a, v2f b, v8f c) {
  return __builtin_amdgcn_wmma_f32_16x16x4_f32(
      /*neg_a=*/false, a, /*neg_b=*/false, b,
      /*c_mod=*/(short)0, c, /*reuse_a=*/false, /*reuse_b=*/false);
}
#else
// Shuffle-based emulation with identical fragment layouts (compile fallback).
static __device__ __forceinline__ v8f wmma4(v2f a, v2f b, v8f c) {
  const int L = threadIdx.x & 31;
  const int half = L >> 4;
#pragma unroll
  for (int k = 0; k < 4; ++k) {
    float bsrc = (k & 1) ? b[1] : b[0];
    float bk = __shfl(bsrc, (L & 15) + ((k >> 1) << 4), 32);
    float asrc = (k & 1) ? a[1] : a[0];
#pragma unroll
    for (int r = 0; r < 8; ++r) {
      float av = __shfl(asrc, (r + (half << 3)) + ((k >> 1) << 4), 32);
      c[r] += av * bk;
    }
  }
  return c;
}
#endif

// ---------------------------------------------------------------------------
// Pack kernels: rearrange U tensors into A-fragment streaming order so each
// wave's A-operand load is one contiguous, fully-coalesced 256B global_load_b64.
// flat dst t = ((mt*KS + ks)*32 + lane)*2 + v ; row = mt*16+(lane&15) = (o,k);
// col m = ks*4 + (lane>>4)*2 + v.
// ---------------------------------------------------------------------------
__global__ void pack3_kernel(const float* __restrict__ U3, float* __restrict__ U3p) {
  unsigned t = blockIdx.x * 256u + threadIdx.x;          // < MT3*KS3*64
  unsigned v = t & 1, lane = (t >> 1) & 31, ks = (t >> 6) & (KS3 - 1), mt = t >> 16;
  unsigned r = mt * 16 + (lane & 15);
  unsigned o = r >> 4, k = r & 15;
  unsigned m = ks * 4 + ((lane >> 4) << 1) + v;          // (i,j,l)
  unsigned i = m >> 8, j = (m >> 4) & 15, l = m & 15;
  U3p[t] = U3[(((o * 16 + i) * 16 + j) * 16 + l) * 16 + k];
}

__global__ void pack2_kernel(const float* __restrict__ U2, float* __restrict__ U2p) {
  unsigned t = blockIdx.x * 256u + threadIdx.x;          // < MT2*KS2*64 = 32768
  unsigned v = t & 1, lane = (t >> 1) & 31, ks = (t >> 6) & (KS2 - 1), mt = t >> 12;
  unsigned r = mt * 16 + (lane & 15);                    // r = o*8 + k2
  unsigned o = r >> 3, k = r & 7;
  unsigned m = ks * 4 + ((lane >> 4) << 1) + v;          // (i,j)
  unsigned i = m >> 4, j = m & 15;
  U2p[t] = U2[((o * 16 + i) * 16 + j) * 8 + k];
}

__global__ void pack1_kernel(const float* __restrict__ U1, float* __restrict__ U1p) {
  unsigned t = blockIdx.x * 256u + threadIdx.x;          // < MT1*KS1*64 = 1024
  unsigned v = t & 1, lane = (t >> 1) & 31, ks = (t >> 6) & (KS1 - 1), mt = t >> 8;
  unsigned r = mt * 16 + (lane & 15);                    // r = o*4 + k1
  unsigned o = r >> 2, k = r & 3;
  unsigned l = ks * 4 + ((lane >> 4) << 1) + v;
  U1p[t] = U1[(o * 16 + l) * 4 + k];
}

// ---------------------------------------------------------------------------
// Main kernel: one workgroup (8 wave32) per 32 (b,c) columns (2 N-tiles).
// ---------------------------------------------------------------------------
__launch_bounds__(256)
__global__ void symcon_main(const float* __restrict__ x,
                            const int*   __restrict__ indices,
                            const float* __restrict__ w1,
                            const float* __restrict__ w2,
                            const float* __restrict__ w3,
                            const float* __restrict__ U3p,
                            const float* __restrict__ U2p,
                            const float* __restrict__ U1p,
                            float* __restrict__ out) {
  __shared__ float xtile[NCOLS * 16];             // x vectors, 32 columns
  __shared__ float Xp[256 * NCOLS];               // monomial panel, B-frag swizzle
  __shared__ float Glds[(256 + 128 + 64) * 16];   // G3|G2|G1, reused per N-tile
  __shared__ int   ecol[NCOLS];                   // element id per column

  const int tid  = threadIdx.x;
  const int lane = tid & 31;
  const int wave = tid >> 5;

  // ---- stage 0: load x tile + element indices (32 cols x 16) ----
#pragma unroll
  for (int q = tid; q < NCOLS * 16; q += 256) {
    int d = q & 15, n = q >> 4;
    int bc = blockIdx.x * NCOLS + n;              // flat (b*64 + c)
    xtile[n * 16 + d] = x[bc * 16 + d];
    if (d == 0) ecol[n] = indices[bc >> 6];       // b = bc / 64
  }
  __syncthreads();

  // Each thread owns column nb for monomial generation (8 threads/column).
  const int nb = tid >> 3;                        // 0..31
  const int sb = tid & 7;
  float xr[16];
#pragma unroll
  for (int d = 0; d < 16; ++d) xr[d] = xtile[nb * 16 + d];

  // B fragments (v2f units): Bp[ks*64 + boff], boff0 for cols 0-15,
  // boff1 (= boff0+32) for cols 16-31.
  const int boff0 = (lane & 15) * 2 + (lane >> 4);
  const int boff1 = boff0 + 32;
  const v2f* Bp = (const v2f*)Xp;

  v8f acc30a = {0.f,0.f,0.f,0.f,0.f,0.f,0.f,0.f};
  v8f acc30b = {0.f,0.f,0.f,0.f,0.f,0.f,0.f,0.f};
  v8f acc31a = {0.f,0.f,0.f,0.f,0.f,0.f,0.f,0.f};
  v8f acc31b = {0.f,0.f,0.f,0.f,0.f,0.f,0.f,0.f};
  v8f acc2a  = {0.f,0.f,0.f,0.f,0.f,0.f,0.f,0.f};
  v8f acc2b  = {0.f,0.f,0.f,0.f,0.f,0.f,0.f,0.f};
  v8f acc1a  = {0.f,0.f,0.f,0.f,0.f,0.f,0.f,0.f};
  v8f acc1b  = {0.f,0.f,0.f,0.f,0.f,0.f,0.f,0.f};

  // ---------------- correlation 3: G3 = U3f[256x4096] @ X3 ----------------
  for (int p = 0; p < 16; ++p) {
    // Build panel of 256 monomials x_i*x_j*x_l for 32 cols, B-frag swizzle:
    // Xp[((mm>>2)*32 + n)*4 + (mm&3)]
#pragma unroll 4
    for (int mm = sb; mm < 256; mm += 8) {
      int m = (p << 8) + mm;
      float val = xr[m >> 8] * xr[(m >> 4) & 15] * xr[m & 15];
      Xp[((mm >> 2) * NCOLS + nb) * 4 + (mm & 3)] = val;
    }
    __syncthreads();

    const v2f* A0 = (const v2f*)U3p + ((size_t)(wave)     * KS3 + p * 64) * 32 + lane;
    const v2f* A1 = (const v2f*)U3p + ((size_t)(wave + 8) * KS3 + p * 64) * 32 + lane;
    __builtin_prefetch(A0 + 64 * 32, 0, 1);     // next K-panel (global_prefetch_b8)
    __builtin_prefetch(A1 + 64 * 32, 0, 1);
#pragma unroll 4
    for (int ks = 0; ks < 64; ++ks) {
      v2f b0 = Bp[ks * 64 + boff0];             // ds_load (compiler pairs them)
      v2f b1 = Bp[ks * 64 + boff1];
      v2f a0 = A0[ks * 32];
      v2f a1 = A1[ks * 32];
      acc30a = wmma4(a0, b0, acc30a);           // v_wmma_f32_16x16x4_f32
      acc30b = wmma4(a0, b1, acc30b);
      acc31a = wmma4(a1, b0, acc31a);
      acc31b = wmma4(a1, b1, acc31b);
    }
    __syncthreads();
  }

  // ---------------- correlation 2: G2 = U2f[128x256] @ X2 -----------------
  {
#pragma unroll 4
    for (int mm = sb; mm < 256; mm += 8) {
      Xp[((mm >> 2) * NCOLS + nb) * 4 + (mm & 3)] = xr[mm >> 4] * xr[mm & 15];
    }
    __syncthreads();
    const v2f* A2 = (const v2f*)U2p + ((size_t)wave * KS2) * 32 + lane;
#pragma unroll 4
    for (int ks = 0; ks < 64; ++ks) {
      v2f b0 = Bp[ks * 64 + boff0];
      v2f b1 = Bp[ks * 64 + boff1];
      v2f a  = A2[ks * 32];
      acc2a = wmma4(a, b0, acc2a);
      acc2b = wmma4(a, b1, acc2b);
    }
    __syncthreads();
  }

  // ---------------- correlation 1: G1 = U1f[64x16] @ X1 -------------------
  {
    if (sb < 2) {                                // mm = sb*8 + {0..7} covers 0..15
#pragma unroll
      for (int mm = sb * 8; mm < sb * 8 + 8; ++mm)
        Xp[((mm >> 2) * NCOLS + nb) * 4 + (mm & 3)] = xr[mm];
    }
    __syncthreads();
    if (wave < 4) {                              // wave-uniform branch
      const v2f* A1p = (const v2f*)U1p + ((size_t)wave * KS1) * 32 + lane;
#pragma unroll
      for (int ks = 0; ks < 4; ++ks) {
        v2f b0 = Bp[ks * 64 + boff0];
        v2f b1 = Bp[ks * 64 + boff1];
        v2f a  = A1p[ks * 32];
        acc1a = wmma4(a, b0, acc1a);
        acc1b = wmma4(a, b1, acc1b);
      }
    }
  }

  // ---- epilogue: for each N-tile, stage G to LDS then weighted-reduce ----
  // D frag: M = r + 8*(lane>>4), N = lane&15.
  const int colb = lane & 15;
  const int mh = (lane >> 4) << 3;
#pragma unroll
  for (int t = 0; t < NT; ++t) {
    __syncthreads();                             // Glds free (prev reduce done)
    {
      v8f a30 = t ? acc30b : acc30a;
      v8f a31 = t ? acc31b : acc31a;
      v8f a2  = t ? acc2b  : acc2a;
#pragma unroll
      for (int r = 0; r < 8; ++r) {
        Glds[(wave * 16 + r + mh) * 16 + colb]       = a30[r];
        Glds[((wave + 8) * 16 + r + mh) * 16 + colb] = a31[r];
        Glds[(256 + wave * 16 + r + mh) * 16 + colb] = a2[r];
      }
      if (wave < 4) {
        v8f a1 = t ? acc1b : acc1a;
#pragma unroll
        for (int r = 0; r < 8; ++r)
          Glds[(384 + wave * 16 + r + mh) * 16 + colb] = a1[r];
      }
    }
    __syncthreads();
    {
      int o = tid & 15;
      int nl = tid >> 4;                         // 0..15 within this N-tile
      int n = t * 16 + nl;
      int bc = blockIdx.x * NCOLS + n;
      int c = bc & 63;
      int e = ecol[n];
      float acc = 0.f;
#pragma unroll
      for (int k = 0; k < 16; ++k)
        acc += w3[(e * 16 + k) * CCH + c] * Glds[(o * 16 + k) * 16 + nl];
#pragma unroll
      for (int k = 0; k < 8; ++k)
        acc += w2[(e * 8 + k) * CCH + c] * Glds[(256 + o * 8 + k) * 16 + nl];
#pragma unroll
      for (int k = 0; k < 4; ++k)
        acc += w1[(e * 4 + k) * CCH + c] * Glds[(384 + o * 4 + k) * 16 + nl];
      out[bc * 16 + o] = acc;                    // contiguous 64B per 16 threads
    }
  }
}

// ---------------------------------------------------------------------------
// Host-side launch. Workspace layout (floats):
//   U3p: MT3*KS3*64 = 1,048,576   (4 MB)
//   U2p: MT2*KS2*64 =    32,768   (128 KB)
//   U1p: MT1*KS1*64 =     1,024   (4 KB)
// ---------------------------------------------------------------------------
extern "C" void kernel_launch(void* const* d_in, const int* in_sizes, int n_in,
                              void* d_out, int out_size, void* d_ws, size_t ws_size,
                              hipStream_t stream) {
  const float* x       = (const float*)d_in[0];
  const int*   indices = (const int*)  d_in[1];
  const float* w1      = (const float*)d_in[2];
  const float* w2      = (const float*)d_in[3];
  const float* w3      = (const float*)d_in[4];
  const float* U1      = (const float*)d_in[5];
  const float* U2      = (const float*)d_in[6];
  const float* U3      = (const float*)d_in[7];
  float* out = (float*)d_out;

  float* U3p = (float*)d_ws;
  float* U2p = U3p + (size_t)MT3 * KS3 * 64;
  float* U1p = U2p + (size_t)MT2 * KS2 * 64;

  // Repack U tensors into WMMA A-fragment streaming order (deterministic,
  // recomputed every call; ~4.2 MB of d_ws).
  pack3_kernel<<<(MT3 * KS3 * 64) / 256, 256, 0, stream>>>(U3, U3p);
  pack2_kernel<<<(MT2 * KS2 * 64) / 256, 256, 0, stream>>>(U2, U2p);
  pack1_kernel<<<(MT1 * KS1 * 64) / 256, 256, 0, stream>>>(U1, U1p);

  const int NC    = in_sizes[0] / NC_D;  // N*C = 65536
  const int tiles = NC / NCOLS;          // 2048 workgroups
  symcon_main<<<tiles, 256, 0, stream>>>(x, indices, w1, w2, w3,
                                         U3p, U2p, U1p, out);
}